// MMCL_83683142795432
// MI455X (gfx1250) — compile-verified
//
#include <hip/hip_runtime.h>
#include <cstdint>
#include <cstddef>

#define HNM_BLOCK 1024
#define HNM_SCALE 10.0f
#define FAST_N    32768
#define FAST_V4   8            // float4 chunks per thread (32 floats)

// Detect gfx1250 async global->LDS builtins (ASYNCcnt path).
#ifdef __has_builtin
#if __has_builtin(__builtin_amdgcn_global_load_async_to_lds_b128)
#define HNM_USE_ASYNC_LDS 1
#endif
#if __has_builtin(__builtin_amdgcn_s_wait_asynccnt)
#define HNM_HAVE_WAIT_ASYNC 1
#endif
#endif

// Builtin parameter type per hipcc diagnostic: vector_size(16) int pointers.
typedef int hnm_v4i __attribute__((vector_size(16)));
typedef __attribute__((address_space(1))) hnm_v4i* hnm_gptr;
typedef __attribute__((address_space(3))) hnm_v4i* hnm_lptr;

__device__ __forceinline__ void hnm_async_wait() {
#ifdef HNM_HAVE_WAIT_ASYNC
    __builtin_amdgcn_s_wait_asynccnt(0);
#else
    asm volatile("s_wait_asynccnt 0" ::: "memory");
#endif
}

// Order-preserving float <-> uint mapping (monotonic for all finite floats).
__device__ __forceinline__ unsigned hnm_f2ord(float x) {
    unsigned b = __float_as_uint(x);
    return (b & 0x80000000u) ? ~b : (b | 0x80000000u);
}
__device__ __forceinline__ float hnm_ord2f(unsigned u) {
    unsigned b = (u & 0x80000000u) ? (u & 0x7fffffffu) : ~u;
    return __uint_as_float(b);
}

// ---------------------------------------------------------------------------
// Fast path: n == 32768, block == 1024. Row is async-copied to LDS once
// (CDNA5 ASYNCcnt path), pulled into VGPRs as ordinals with ONE ds_load_b128
// sweep, and all passes (max, 4x8-bit radix select, exp-sum) run from
// registers. LDS carries only the stage + one read + 256-bin histograms:
// ~260KB/WG, far below the HBM roofline (512MB @ 23.3TB/s ~ 22us).
// ---------------------------------------------------------------------------
__global__ __launch_bounds__(HNM_BLOCK)
void hnm_row_fast(const float* __restrict__ inputs,
                  const long long* __restrict__ targets,
                  float* __restrict__ row_loss, int k) {
    extern __shared__ float srow[];          // FAST_N floats (128 KB)
    __shared__ unsigned hist[256];
    __shared__ float    redf[HNM_BLOCK];
    __shared__ unsigned redu[HNM_BLOCK];
    __shared__ unsigned s_prefix;
    __shared__ unsigned s_rem;
    __shared__ float    s_pos;
    __shared__ float    s_max;

    const int row = blockIdx.x;
    const int tid = threadIdx.x;
    const int tgt = (int)targets[row];
    const float* grow = inputs + (size_t)row * (size_t)FAST_N;

    // ---- Stage row into LDS via async global->LDS (b128 per lane) ----
#ifdef HNM_USE_ASYNC_LDS
#pragma unroll
    for (int i = 0; i < FAST_V4; ++i) {
        const int c = tid + i * HNM_BLOCK;   // float4 chunk index
        __builtin_amdgcn_global_load_async_to_lds_b128(
            (hnm_gptr)(grow + 4 * c), (hnm_lptr)(srow + 4 * c), 0, 0);
    }
    hnm_async_wait();
#else
    {
        const float4* g4 = reinterpret_cast<const float4*>(grow);
        float4*       s4 = reinterpret_cast<float4*>(srow);
#pragma unroll
        for (int i = 0; i < FAST_V4; ++i) s4[tid + i * HNM_BLOCK] = g4[tid + i * HNM_BLOCK];
    }
#endif
    __syncthreads();

    // ---- Single LDS sweep: fill registers with ordinals, strip positive ----
    unsigned uu[FAST_V4 * 4];
    unsigned umax = 0u;
    {
        const float4* s4 = reinterpret_cast<const float4*>(srow);
#pragma unroll
        for (int i = 0; i < FAST_V4; ++i) {
            const int c  = tid + i * HNM_BLOCK;
            const int j0 = 4 * c;
            float4 v = s4[c];                          // ds_load_b128
            float xs0 = v.x, xs1 = v.y, xs2 = v.z, xs3 = v.w;
            unsigned u0 = hnm_f2ord(xs0), u1 = hnm_f2ord(xs1);
            unsigned u2 = hnm_f2ord(xs2), u3 = hnm_f2ord(xs3);
            if (j0 + 0 == tgt) { s_pos = xs0; u0 = 0u; }
            if (j0 + 1 == tgt) { s_pos = xs1; u1 = 0u; }
            if (j0 + 2 == tgt) { s_pos = xs2; u2 = 0u; }
            if (j0 + 3 == tgt) { s_pos = xs3; u3 = 0u; }
            uu[4 * i + 0] = u0; uu[4 * i + 1] = u1;
            uu[4 * i + 2] = u2; uu[4 * i + 3] = u3;
            unsigned a = u0 > u1 ? u0 : u1;
            unsigned b = u2 > u3 ? u2 : u3;
            unsigned c2 = a > b ? a : b;
            umax = umax > c2 ? umax : c2;
        }
    }
    redu[tid] = umax;
    __syncthreads();
    for (int s = HNM_BLOCK >> 1; s > 0; s >>= 1) {
        if (tid < s) redu[tid] = redu[tid] > redu[tid + s] ? redu[tid] : redu[tid + s];
        __syncthreads();
    }
    if (tid == 0) {
        s_max    = fmaxf(hnm_ord2f(redu[0]), s_pos);   // M attained in kept set
        s_prefix = 0u;
        s_rem    = (unsigned)k;
    }
    __syncthreads();
    const float M = s_max;

    // ---- Exact k-th largest negative: 4-pass radix from registers ----
    for (int p = 0; p < 4; ++p) {
        const int      shift     = 24 - 8 * p;
        const unsigned maskAbove = (p == 0) ? 0u : (0xFFFFFFFFu << (shift + 8));
        if (tid < 256) hist[tid] = 0u;
        __syncthreads();
        const unsigned pref = s_prefix;
#pragma unroll
        for (int e = 0; e < FAST_V4 * 4; ++e) {
            const unsigned u = uu[e];
            if ((u & maskAbove) == pref)
                atomicAdd(&hist[(u >> shift) & 0xFFu], 1u);
        }
        __syncthreads();
        // Parallel inclusive suffix scan over 256 bins (Hillis-Steele).
        for (int d = 1; d < 256; d <<= 1) {
            unsigned t = 0u;
            if (tid < 256) t = (tid + d < 256) ? hist[tid + d] : 0u;
            __syncthreads();
            if (tid < 256) hist[tid] += t;
            __syncthreads();
        }
        const unsigned rem = s_rem;
        __syncthreads();
        if (tid < 256) {
            const unsigned S     = hist[tid];                     // count >= bin tid
            const unsigned Snext = (tid == 255) ? 0u : hist[tid + 1];
            if (S >= rem && Snext < rem) {                        // unique winner
                s_prefix = pref | ((unsigned)tid << shift);
                s_rem    = rem - Snext;
            }
        }
        __syncthreads();
    }
    const unsigned tu = s_prefix;            // ordinal of the k-th largest

    // ---- exp-sum over strict-greater negatives (from registers) ----
    float    lsum = 0.0f;
    unsigned lcnt = 0u;
#pragma unroll
    for (int e = 0; e < FAST_V4 * 4; ++e) {
        const unsigned u = uu[e];
        if (u > tu) {
            lsum += __expf(HNM_SCALE * (hnm_ord2f(u) - M));
            lcnt += 1u;
        }
    }
    redf[tid] = lsum; redu[tid] = lcnt;
    __syncthreads();
    for (int s = HNM_BLOCK >> 1; s > 0; s >>= 1) {
        if (tid < s) { redf[tid] += redf[tid + s]; redu[tid] += redu[tid + s]; }
        __syncthreads();
    }
    if (tid == 0) {
        const float pos  = s_pos;
        const float tval = hnm_ord2f(tu);
        float total = redf[0]
                    + (float)(k - (int)redu[0]) * __expf(HNM_SCALE * (tval - M))
                    + __expf(HNM_SCALE * (pos - M));
        row_loss[row] = __logf(total) + HNM_SCALE * M - HNM_SCALE * pos;
    }
}

// ---------------------------------------------------------------------------
// Generic fallback (any n, n staged in LDS, scans from LDS). Same algorithm.
// ---------------------------------------------------------------------------
__global__ __launch_bounds__(HNM_BLOCK)
void hnm_row_generic(const float* __restrict__ inputs,
                     const long long* __restrict__ targets,
                     float* __restrict__ row_loss,
                     int n, int k) {
    extern __shared__ float srow[];
    __shared__ unsigned hist[256];
    __shared__ float    redf[HNM_BLOCK];
    __shared__ unsigned redu[HNM_BLOCK];
    __shared__ unsigned s_prefix;
    __shared__ unsigned s_rem;
    __shared__ float    s_max;

    const int  row  = blockIdx.x;
    const int  tid  = threadIdx.x;
    const int  nthr = blockDim.x;
    const int  tgt  = (int)targets[row];
    const float* grow = inputs + (size_t)row * (size_t)n;

    {
        const int nvec = n >> 2;
#ifdef HNM_USE_ASYNC_LDS
        for (int i = tid; i < nvec; i += nthr) {
            __builtin_amdgcn_global_load_async_to_lds_b128(
                (hnm_gptr)(grow + 4 * i), (hnm_lptr)(srow + 4 * i), 0, 0);
        }
        hnm_async_wait();
#else
        const float4* g4 = reinterpret_cast<const float4*>(grow);
        float4*       s4 = reinterpret_cast<float4*>(srow);
        for (int i = tid; i < nvec; i += nthr) s4[i] = g4[i];
#endif
        for (int j = 4 * nvec + tid; j < n; j += nthr) srow[j] = grow[j]; // tail
    }
    __syncthreads();

    float lmax = -3.4028234663852886e38f;
    for (int j = tid; j < n; j += nthr) lmax = fmaxf(lmax, srow[j]);
    redf[tid] = lmax;
    __syncthreads();
    for (int s = nthr >> 1; s > 0; s >>= 1) {
        if (tid < s) redf[tid] = fmaxf(redf[tid], redf[tid + s]);
        __syncthreads();
    }
    if (tid == 0) { s_max = redf[0]; s_prefix = 0u; s_rem = (unsigned)k; }
    __syncthreads();
    const float M = s_max;

    for (int p = 0; p < 4; ++p) {
        const int      shift     = 24 - 8 * p;
        const unsigned maskAbove = (p == 0) ? 0u : (0xFFFFFFFFu << (shift + 8));
        if (tid < 256) hist[tid] = 0u;
        __syncthreads();
        const unsigned pref = s_prefix;
        for (int j = tid; j < n; j += nthr) {
            if (j == tgt) continue;
            unsigned u = hnm_f2ord(srow[j]);
            if ((u & maskAbove) == pref)
                atomicAdd(&hist[(u >> shift) & 0xFFu], 1u);
        }
        __syncthreads();
        for (int d = 1; d < 256; d <<= 1) {
            unsigned t = 0u;
            if (tid < 256) t = (tid + d < 256) ? hist[tid + d] : 0u;
            __syncthreads();
            if (tid < 256) hist[tid] += t;
            __syncthreads();
        }
        const unsigned rem = s_rem;
        __syncthreads();
        if (tid < 256) {
            const unsigned S     = hist[tid];
            const unsigned Snext = (tid == 255) ? 0u : hist[tid + 1];
            if (S >= rem && Snext < rem) {
                s_prefix = pref | ((unsigned)tid << shift);
                s_rem    = rem - Snext;
            }
        }
        __syncthreads();
    }
    const unsigned tu = s_prefix;

    float    lsum = 0.0f;
    unsigned lcnt = 0u;
    for (int j = tid; j < n; j += nthr) {
        if (j == tgt) continue;
        float x = srow[j];
        if (hnm_f2ord(x) > tu) { lsum += __expf(HNM_SCALE * (x - M)); lcnt += 1u; }
    }
    redf[tid] = lsum; redu[tid] = lcnt;
    __syncthreads();
    for (int s = nthr >> 1; s > 0; s >>= 1) {
        if (tid < s) { redf[tid] += redf[tid + s]; redu[tid] += redu[tid + s]; }
        __syncthreads();
    }
    if (tid == 0) {
        const float pos  = srow[tgt];
        const float tval = hnm_ord2f(tu);
        float total = redf[0]
                    + (float)(k - (int)redu[0]) * __expf(HNM_SCALE * (tval - M))
                    + __expf(HNM_SCALE * (pos - M));
        row_loss[row] = __logf(total) + HNM_SCALE * M - HNM_SCALE * pos;
    }
}

// Deterministic single-block tree reduction: mean of m per-row losses.
__global__ __launch_bounds__(1024)
void hnm_reduce_kernel(const float* __restrict__ row_loss,
                       float* __restrict__ out, int m) {
    __shared__ float red[1024];
    const int tid = threadIdx.x;
    float s = 0.0f;
    for (int i = tid; i < m; i += blockDim.x) s += row_loss[i];
    red[tid] = s;
    __syncthreads();
    for (int st = blockDim.x >> 1; st > 0; st >>= 1) {
        if (tid < st) red[tid] += red[tid + st];
        __syncthreads();
    }
    if (tid == 0) out[0] = red[0] / (float)m;
}

extern "C" void kernel_launch(void* const* d_in, const int* in_sizes, int n_in,
                              void* d_out, int out_size, void* d_ws, size_t ws_size,
                              hipStream_t stream) {
    const float*     inputs  = (const float*)d_in[0];
    const long long* targets = (const long long*)d_in[1];  // int64 targets
    const int m = in_sizes[1];
    const int n = in_sizes[0] / m;
    const int k = (int)(0.01 * (double)(n - 1));           // 327 for n=32768

    float* row_loss = (float*)d_ws;                        // m floats scratch
    const size_t shmem = (size_t)n * sizeof(float);        // row stage in LDS

    if (n == FAST_N) {
        (void)hipFuncSetAttribute((const void*)hnm_row_fast,
                                  hipFuncAttributeMaxDynamicSharedMemorySize,
                                  (int)shmem);
        hipLaunchKernelGGL(hnm_row_fast, dim3(m), dim3(HNM_BLOCK), shmem, stream,
                           inputs, targets, row_loss, k);
    } else {
        (void)hipFuncSetAttribute((const void*)hnm_row_generic,
                                  hipFuncAttributeMaxDynamicSharedMemorySize,
                                  (int)shmem);
        hipLaunchKernelGGL(hnm_row_generic, dim3(m), dim3(HNM_BLOCK), shmem, stream,
                           inputs, targets, row_loss, n, k);
    }
    hipLaunchKernelGGL(hnm_reduce_kernel, dim3(1), dim3(1024), 0, stream,
                       row_loss, (float*)d_out, m);
}